// PopulationODE_13185549598961
// MI455X (gfx1250) — compile-verified
//
#include <hip/hip_runtime.h>

// ---------------------------------------------------------------------------
// PopulationODE: sequential RK4 scan over a 32-wide MLP vector field.
// Single wave32, weights in registers (WMMA A-layout), f32 WMMA 16x16x4.
// ---------------------------------------------------------------------------

typedef __attribute__((ext_vector_type(2))) float v2f;
typedef __attribute__((ext_vector_type(8))) float v8f;

#define WMMA_F32(a, b, c) \
  __builtin_amdgcn_wmma_f32_16x16x4_f32(false, (a), false, (b), (short)0, (c), false, false)

__device__ __forceinline__ float sigmoidf_(float v) {
  return 1.0f / (1.0f + expf(-v));
}

// 32x32 f32 matvec: D(32) = W(32x32) @ x(32), x replicated across all 16 B
// columns.  w = A-operand tiles [mtile][ktile].  xb = x in LDS (32 floats).
// hi2 = (lane>=16) ? 2 : 0.  Two independent accumulator chains per M-tile.
__device__ __forceinline__ void mm32(const v2f (&w)[2][8],
                                     const float* xb, int hi2,
                                     v8f& d0, v8f& d1) {
  v8f c00 = {0.f,0.f,0.f,0.f,0.f,0.f,0.f,0.f};
  v8f c01 = c00, c10 = c00, c11 = c00;
#pragma unroll
  for (int k = 0; k < 4; ++k) {
    v2f b = *(const v2f*)(xb + 4 * k + hi2);   // rows k0..k0+3 of B = x slice
    c00 = WMMA_F32(w[0][k], b, c00);
    c10 = WMMA_F32(w[1][k], b, c10);
  }
#pragma unroll
  for (int k = 4; k < 8; ++k) {
    v2f b = *(const v2f*)(xb + 4 * k + hi2);
    c01 = WMMA_F32(w[0][k], b, c01);
    c11 = WMMA_F32(w[1][k], b, c11);
  }
  d0 = c00 + c01;
  d1 = c10 + c11;
}

// D (column-replicated) -> 32 floats in LDS.  Lane 0 holds rows 0..7 (d0) and
// 16..23 (d1); lane 16 holds rows 8..15 (d0) and 24..31 (d1).
__device__ __forceinline__ void scatter32(v8f d0, v8f d1, float* buf, int lane) {
  if (lane == 0) {
#pragma unroll
    for (int r = 0; r < 8; ++r) { buf[r]      = d0[r]; buf[16 + r] = d1[r]; }
  } else if (lane == 16) {
#pragma unroll
    for (int r = 0; r < 8; ++r) { buf[8 + r]  = d0[r]; buf[24 + r] = d1[r]; }
  }
  __syncthreads();   // single-wave workgroup: barrier is ~free (ISA 3.1)
}

__global__ void __launch_bounds__(32)
pop_ode_kernel(const float* __restrict__ s_grid, const float* __restrict__ y0v,
               const float* __restrict__ Win,  const float* __restrict__ binp,
               const float* __restrict__ W1a,  const float* __restrict__ b1a,
               const float* __restrict__ W1b,  const float* __restrict__ b1b,
               const float* __restrict__ W2a,  const float* __restrict__ b2a,
               const float* __restrict__ W2b,  const float* __restrict__ b2b,
               const float* __restrict__ Wout, const float* __restrict__ boutv,
               const float* __restrict__ Av,
               float* __restrict__ out, int T) {
  const int lane = threadIdx.x;       // 0..31 (one wave32)
  const int rlo  = lane & 15;         // row within 16-row tile
  const int hi   = lane >> 4;         // lane half
  const int hi2  = hi * 2;

  __shared__ float xb[32];            // canonical activation vector
  __shared__ float tb[32];            // matvec result staging
  __shared__ float hb[32];            // inner tanh staging
  __shared__ float fb[4];             // final 4-vector staging

  // ---- Load 32x32 weights into WMMA A-operand layout (registers) ----------
  v2f w1a[2][8], w1b[2][8], w2a[2][8], w2b[2][8];
#pragma unroll
  for (int mt = 0; mt < 2; ++mt) {
#pragma unroll
    for (int kt = 0; kt < 8; ++kt) {
      const int row = mt * 16 + rlo;
      const int col = kt * 4 + hi2;                 // even -> 8B aligned
      w1a[mt][kt] = *(const v2f*)(W1a + row * 32 + col);
      w1b[mt][kt] = *(const v2f*)(W1b + row * 32 + col);
      w2a[mt][kt] = *(const v2f*)(W2a + row * 32 + col);
      w2b[mt][kt] = *(const v2f*)(W2b + row * 32 + col);
    }
  }

  // Win (32x5) padded to 32x8: 2 M-tiles x 2 K-tiles.
  v2f win[2][2];
#pragma unroll
  for (int mt = 0; mt < 2; ++mt) {
    const int row = mt * 16 + rlo;
    v2f a0; a0.x = Win[row * 5 + hi2]; a0.y = Win[row * 5 + hi2 + 1]; // cols 0..3
    win[mt][0] = a0;
    v2f a1; a1.x = (hi == 0) ? Win[row * 5 + 4] : 0.0f; a1.y = 0.0f; // col 4, pad
    win[mt][1] = a1;
  }

  // Wout (4x32) padded to 16x32: 1 M-tile x 8 K-tiles (rows >=4 are zero).
  v2f wo[8];
#pragma unroll
  for (int kt = 0; kt < 8; ++kt) {
    const int col = kt * 4 + hi2;
    v2f a; a.x = 0.0f; a.y = 0.0f;
    if (rlo < 4) a = *(const v2f*)(Wout + rlo * 32 + col);
    wo[kt] = a;
  }

  // Per-lane biases; uniform small vectors.
  const float binl = binp[lane];
  const float b1al = b1a[lane], b1bl = b1b[lane];
  const float b2al = b2a[lane], b2bl = b2b[lane];
  const float bo0 = boutv[0], bo1 = boutv[1], bo2 = boutv[2], bo3 = boutv[3];
  const float A0 = Av[0], A1 = Av[1], A2 = Av[2], A3 = Av[3];

  float ya = y0v[0], yb = y0v[1], yc = y0v[2], yd = y0v[3];
  if (lane < 4) out[lane] = y0v[lane];   // out[0,:] = y0

  // ---- vector field f(z, s) -> 4 uniform floats ---------------------------
  auto feval = [&](float za, float zb, float zc, float zd, float s,
                   float& f0, float& f1, float& f2, float& f3) {
    v8f d0, d1;
    // x = tanh(Win @ [z, s] + bin): z is uniform, build B directly.
    {
      v2f bv0, bv1;
      if (hi) { bv0.x = zc; bv0.y = zd; bv1.x = 0.f; bv1.y = 0.f; }
      else    { bv0.x = za; bv0.y = zb; bv1.x = s;   bv1.y = 0.f; }
      v8f c0 = {0.f,0.f,0.f,0.f,0.f,0.f,0.f,0.f};
      v8f c1 = c0;
      c0 = WMMA_F32(win[0][0], bv0, c0);
      c1 = WMMA_F32(win[1][0], bv0, c1);
      c0 = WMMA_F32(win[0][1], bv1, c0);
      c1 = WMMA_F32(win[1][1], bv1, c1);
      d0 = c0; d1 = c1;
    }
    scatter32(d0, d1, tb, lane);
    float x = tanhf(tb[lane] + binl);
    xb[lane] = x;
    __syncthreads();

    // Residual block 1: x += W1b @ tanh(W1a @ x + b1a) + b1b
    mm32(w1a, xb, hi2, d0, d1);
    scatter32(d0, d1, tb, lane);
    hb[lane] = tanhf(tb[lane] + b1al);
    __syncthreads();
    mm32(w1b, hb, hi2, d0, d1);
    scatter32(d0, d1, tb, lane);
    x = x + tb[lane] + b1bl;
    xb[lane] = x;
    __syncthreads();

    // Residual block 2
    mm32(w2a, xb, hi2, d0, d1);
    scatter32(d0, d1, tb, lane);
    hb[lane] = tanhf(tb[lane] + b2al);
    __syncthreads();
    mm32(w2b, hb, hi2, d0, d1);
    scatter32(d0, d1, tb, lane);
    x = x + tb[lane] + b2bl;
    xb[lane] = x;
    __syncthreads();

    // net = sigmoid(Wout @ x + bout): single M-tile, 8 K-tiles.
    {
      v8f ca = {0.f,0.f,0.f,0.f,0.f,0.f,0.f,0.f};
      v8f cb = ca;
#pragma unroll
      for (int k = 0; k < 4; ++k) {
        v2f bv = *(const v2f*)(xb + 4 * k + hi2);
        ca = WMMA_F32(wo[k], bv, ca);
      }
#pragma unroll
      for (int k = 4; k < 8; ++k) {
        v2f bv = *(const v2f*)(xb + 4 * k + hi2);
        cb = WMMA_F32(wo[k], bv, cb);
      }
      v8f dF = ca + cb;               // rows 0..3 live in lanes 0..15
      if (lane == 0) { fb[0] = dF[0]; fb[1] = dF[1]; fb[2] = dF[2]; fb[3] = dF[3]; }
      __syncthreads();
    }
    const float n0 = sigmoidf_(fb[0] + bo0);
    const float n1 = sigmoidf_(fb[1] + bo1);
    const float n2 = sigmoidf_(fb[2] + bo2);
    const float n3 = sigmoidf_(fb[3] + bo3);
    __syncthreads();

    const float s0 = sigmoidf_(za), s1 = sigmoidf_(zb);
    const float s2 = sigmoidf_(zc), s3 = sigmoidf_(zd);
    f0 = 0.01f * s0 * (A0 - s0) * n0;
    f1 = 0.01f * s1 * (A1 - s1) * n1;
    f2 = 0.01f * s2 * (A2 - s2) * n2;
    f3 = 0.01f * s3 * (A3 - s3) * n3;
  };

  // ---- RK4 scan -----------------------------------------------------------
  const int steps = T - 1;
  for (int t = 0; t < steps; ++t) {
    const float sp = s_grid[t];
    const float h  = s_grid[t + 1] - sp;

    float k10, k11, k12, k13, k20, k21, k22, k23;
    float k30, k31, k32, k33, k40, k41, k42, k43;

    feval(ya, yb, yc, yd, sp, k10, k11, k12, k13);
    feval(ya + 0.5f * h * k10, yb + 0.5f * h * k11,
          yc + 0.5f * h * k12, yd + 0.5f * h * k13, sp, k20, k21, k22, k23);
    feval(ya + 0.5f * h * k20, yb + 0.5f * h * k21,
          yc + 0.5f * h * k22, yd + 0.5f * h * k23, sp, k30, k31, k32, k33);
    feval(ya + h * k30, yb + h * k31,
          yc + h * k32, yd + h * k33, sp, k40, k41, k42, k43);

    const float w6 = h * (1.0f / 6.0f);
    ya += w6 * (k10 + 2.0f * k20 + 2.0f * k30 + k40);
    yb += w6 * (k11 + 2.0f * k21 + 2.0f * k31 + k41);
    yc += w6 * (k12 + 2.0f * k22 + 2.0f * k32 + k42);
    yd += w6 * (k13 + 2.0f * k23 + 2.0f * k33 + k43);

    if (lane < 4) {
      const float v = (lane == 0) ? ya : (lane == 1) ? yb : (lane == 2) ? yc : yd;
      out[(size_t)(t + 1) * 4 + lane] = v;
    }
  }
}

extern "C" void kernel_launch(void* const* d_in, const int* in_sizes, int n_in,
                              void* d_out, int out_size, void* d_ws, size_t ws_size,
                              hipStream_t stream) {
  const float* s_grid = (const float*)d_in[0];
  const float* y0v    = (const float*)d_in[1];
  const float* Win    = (const float*)d_in[2];
  const float* binp   = (const float*)d_in[3];
  const float* W1a    = (const float*)d_in[4];
  const float* b1a    = (const float*)d_in[5];
  const float* W1b    = (const float*)d_in[6];
  const float* b1b    = (const float*)d_in[7];
  const float* W2a    = (const float*)d_in[8];
  const float* b2a    = (const float*)d_in[9];
  const float* W2b    = (const float*)d_in[10];
  const float* b2b    = (const float*)d_in[11];
  const float* Wout   = (const float*)d_in[12];
  const float* boutv  = (const float*)d_in[13];
  const float* Av     = (const float*)d_in[14];
  const int T = in_sizes[0];

  // Strictly sequential scan: one workgroup, one wave32 owns the whole chain.
  pop_ode_kernel<<<1, 32, 0, stream>>>(s_grid, y0v, Win, binp, W1a, b1a, W1b, b1b,
                                       W2a, b2a, W2b, b2b, Wout, boutv, Av,
                                       (float*)d_out, T);
}